// MIPNetwork_45827301048617
// MI455X (gfx1250) — compile-verified
//
#include <hip/hip_runtime.h>
#include <hip/hip_bf16.h>
#include <math.h>

#define FV       16
#define VCOUNT   500000
#define CCOUNT   250000
#define NEDGES   8000000

typedef __attribute__((ext_vector_type(2))) float v2f;
typedef __attribute__((ext_vector_type(8))) float v8f;

// ---------------------------------------------------------------------------
// init: variables = 1.0, constraints = 0.0
// ---------------------------------------------------------------------------
__global__ void __launch_bounds__(256) k_init(float* __restrict__ vars,
                                              float* __restrict__ cons) {
    size_t i = (size_t)blockIdx.x * 256 + threadIdx.x;
    if (i < (size_t)VCOUNT * FV) vars[i] = 1.0f;
    if (i < (size_t)CCOUNT * FV) cons[i] = 0.0f;
}

// ---------------------------------------------------------------------------
// zero an accumulator (float4 granularity; n4 = count/4)
// ---------------------------------------------------------------------------
__global__ void __launch_bounds__(256) k_zero4(float4* __restrict__ p, int n4) {
    int i = blockIdx.x * 256 + threadIdx.x;
    if (i < n4) p[i] = make_float4(0.f, 0.f, 0.f, 0.f);
}

// ---------------------------------------------------------------------------
// edge scatter: acc[sidx[e]] += vals[e] * src[gidx[e]]   (16 f32 atomics/edge)
// Node arrays are L2-resident (<=48MB); atomics are no-return (STOREcnt).
// ---------------------------------------------------------------------------
__global__ void __launch_bounds__(256) k_edge_scatter(
    const int*   __restrict__ gidx,
    const int*   __restrict__ sidx,
    const float* __restrict__ vals,
    const float* __restrict__ src,
    float*       __restrict__ acc,
    int n)
{
    int e = blockIdx.x * 256 + threadIdx.x;
    if (e >= n) return;
    // stream-ahead prefetch of the edge arrays (one lane per cacheline group)
    if ((threadIdx.x & 31) == 0) {
        __builtin_prefetch(gidx + e + 8192, 0, 0);
        __builtin_prefetch(sidx + e + 8192, 0, 0);
        __builtin_prefetch(vals + e + 8192, 0, 0);
    }
    int   g = gidx[e];
    int   s = sidx[e];
    float v = vals[e];
    const float4* sp = (const float4*)(src + (size_t)g * FV);
    float4 x0 = sp[0], x1 = sp[1], x2 = sp[2], x3 = sp[3];
    float* ap = acc + (size_t)s * FV;
    atomicAdd(ap +  0, v * x0.x); atomicAdd(ap +  1, v * x0.y);
    atomicAdd(ap +  2, v * x0.z); atomicAdd(ap +  3, v * x0.w);
    atomicAdd(ap +  4, v * x1.x); atomicAdd(ap +  5, v * x1.y);
    atomicAdd(ap +  6, v * x1.z); atomicAdd(ap +  7, v * x1.w);
    atomicAdd(ap +  8, v * x2.x); atomicAdd(ap +  9, v * x2.y);
    atomicAdd(ap + 10, v * x2.z); atomicAdd(ap + 11, v * x2.w);
    atomicAdd(ap + 12, v * x3.x); atomicAdd(ap + 13, v * x3.y);
    atomicAdd(ap + 14, v * x3.z); atomicAdd(ap + 15, v * x3.w);
}

// ---------------------------------------------------------------------------
// out[n,16] = relu( [src0 | src1][n,32] @ W[32,16] + bias )
// One 16-node tile per wave32; 8x V_WMMA_F32_16X16X4_F32 (full f32 precision).
// A lane layout (16x4):  lane -> (M = lane&15, K = k0 + 2*(lane>>4) + v)
// B lane layout (4x16):  lane -> (K = k0 + 2*(lane>>4) + v, N = lane&15)
// C/D (16x16):           reg r -> (M = r + 8*(lane>>4), N = lane&15)
// ---------------------------------------------------------------------------
__global__ void __launch_bounds__(256) k_linear32(
    const float* __restrict__ src0,   // old node features [ntiles*16, 16]
    const float* __restrict__ src1,   // aggregated messages [ntiles*16, 16]
    const float* __restrict__ W,      // [32,16] row-major (in,out)
    const float* __restrict__ bias,   // [16]
    float*       __restrict__ out,    // [ntiles*16, 16] (may alias src0)
    int ntiles)
{
    int lane = threadIdx.x & 31;
    int wave = threadIdx.x >> 5;
    int tile = blockIdx.x * 8 + wave;
    if (tile >= ntiles) return;          // wave-uniform: EXEC all-1s below
    int m    = lane & 15;
    int half = lane >> 4;

    v2f b[8];
#pragma unroll
    for (int k = 0; k < 8; ++k) {
        int kr = k * 4 + half * 2;
        b[k].x = W[(kr + 0) * FV + m];
        b[k].y = W[(kr + 1) * FV + m];
    }
    float bm = bias[m];
    v8f c;
#pragma unroll
    for (int r = 0; r < 8; ++r) c[r] = bm;

    const float* row0 = src0 + ((size_t)tile * 16 + m) * FV;
    const float* row1 = src1 + ((size_t)tile * 16 + m) * FV;
#pragma unroll
    for (int k = 0; k < 8; ++k) {
        const float* rp = (k < 4) ? row0 : row1;
        int koff = (k & 3) * 4 + half * 2;
        v2f a = *(const v2f*)(rp + koff);
        c = __builtin_amdgcn_wmma_f32_16x16x4_f32(
                /*neg_a=*/false, a, /*neg_b=*/false, b[k],
                /*c_mod=*/(short)0, c, /*reuse_a=*/false, /*reuse_b=*/false);
    }

    float* op = out + (size_t)tile * 16 * FV;
#pragma unroll
    for (int r = 0; r < 8; ++r) {
        float vv = c[r] > 0.f ? c[r] : 0.f;
        op[(size_t)(r + 8 * half) * FV + m] = vv;   // in-place safe: tile-private
    }
}

// ---------------------------------------------------------------------------
// deterministic N(0,1) per output row (PCG hash + Box-Muller)
// ---------------------------------------------------------------------------
__device__ __forceinline__ unsigned pcg_hash(unsigned x) {
    x = x * 747796405u + 2891336453u;
    unsigned w = ((x >> ((x >> 28) + 4u)) ^ x) * 277803737u;
    return (w >> 22) ^ w;
}
__device__ __forceinline__ float gauss01(unsigned i) {
    unsigned h1 = pcg_hash(i * 2u + 0x9E3779B9u);
    unsigned h2 = pcg_hash(i * 2u + 0x85EBCA6Bu);
    float u1 = (float)(h1 | 1u) * 2.3283064365386963e-10f;  // (0,1]
    float u2 = (float)h2        * 2.3283064365386963e-10f;  // [0,1)
    return sqrtf(-2.f * logf(u1)) * cosf(6.283185307179586f * u2);
}

// ---------------------------------------------------------------------------
// head: h = relu(vars@Wo + bo) [16x16 via 4 WMMA]; a = h@Wo2 + bo2 + 8*N(0,1);
// out = sigmoid(a).  Cross-lane reduction over N via wave32 shuffles.
// ---------------------------------------------------------------------------
__global__ void __launch_bounds__(256) k_output(
    const float* __restrict__ vars,  // [VCOUNT,16]
    const float* __restrict__ Wo,    // [16,16]
    const float* __restrict__ bo,    // [16]
    const float* __restrict__ Wo2,   // [16,1]
    const float* __restrict__ bo2,   // [1]
    float*       __restrict__ out,   // [VCOUNT]
    int ntiles)
{
    int lane = threadIdx.x & 31;
    int wave = threadIdx.x >> 5;
    int tile = blockIdx.x * 8 + wave;
    if (tile >= ntiles) return;          // wave-uniform
    int m    = lane & 15;
    int half = lane >> 4;

    v2f b[4];
#pragma unroll
    for (int k = 0; k < 4; ++k) {
        int kr = k * 4 + half * 2;
        b[k].x = Wo[(kr + 0) * FV + m];
        b[k].y = Wo[(kr + 1) * FV + m];
    }
    float bm = bo[m];
    v8f c;
#pragma unroll
    for (int r = 0; r < 8; ++r) c[r] = bm;

    const float* rp = vars + ((size_t)tile * 16 + m) * FV;
#pragma unroll
    for (int k = 0; k < 4; ++k) {
        v2f a = *(const v2f*)(rp + k * 4 + half * 2);
        c = __builtin_amdgcn_wmma_f32_16x16x4_f32(
                false, a, false, b[k], (short)0, c, false, false);
    }

    // second projection: a[M] = sum_N relu(h[M][N]) * Wo2[N]
    float w2 = Wo2[m];
    float a8[8];
#pragma unroll
    for (int r = 0; r < 8; ++r) {
        float p = (c[r] > 0.f ? c[r] : 0.f) * w2;
        p += __shfl_xor(p, 1, 32);
        p += __shfl_xor(p, 2, 32);
        p += __shfl_xor(p, 4, 32);
        p += __shfl_xor(p, 8, 32);       // reduce within each 16-lane half
        a8[r] = p;
    }
    if (m == 0) {
        float b2 = bo2[0];
#pragma unroll
        for (int r = 0; r < 8; ++r) {
            int row = tile * 16 + r + 8 * half;
            float x = a8[r] + b2 + 8.0f * gauss01((unsigned)row);
            out[row] = 1.0f / (1.0f + expf(-x));
        }
    }
}

// ---------------------------------------------------------------------------
// launcher
// ---------------------------------------------------------------------------
extern "C" void kernel_launch(void* const* d_in, const int* in_sizes, int n_in,
                              void* d_out, int out_size, void* d_ws, size_t ws_size,
                              hipStream_t stream) {
    const int*   rows = (const int*)  d_in[0];
    const int*   cols = (const int*)  d_in[1];
    const float* vals = (const float*)d_in[2];
    // d_in[3] (conditions_values) is unused by the reference computation.
    const float* Wc  = (const float*)d_in[4];
    const float* bc  = (const float*)d_in[5];
    const float* Wv  = (const float*)d_in[6];
    const float* bv  = (const float*)d_in[7];
    const float* Wo  = (const float*)d_in[8];
    const float* bo  = (const float*)d_in[9];
    const float* Wo2 = (const float*)d_in[10];
    const float* bo2 = (const float*)d_in[11];
    float* outp = (float*)d_out;

    // workspace layout (96 MB of f32; L2-resident node state)
    float* ws   = (float*)d_ws;
    float* vars = ws;                                  // VCOUNT*16
    float* cons = vars + (size_t)VCOUNT * FV;          // CCOUNT*16
    float* vacc = cons + (size_t)CCOUNT * FV;          // VCOUNT*16
    float* cacc = vacc + (size_t)VCOUNT * FV;          // CCOUNT*16

    const int initN   = VCOUNT * FV;                   // covers cons too
    const int edgeBlk = (NEDGES + 255) / 256;
    const int ctiles  = CCOUNT / 16;
    const int vtiles  = VCOUNT / 16;

    k_init<<<(initN + 255) / 256, 256, 0, stream>>>(vars, cons);

    for (int pass = 0; pass < 2; ++pass) {
        // v -> c
        k_zero4<<<(CCOUNT * FV / 4 + 255) / 256, 256, 0, stream>>>(
            (float4*)cacc, CCOUNT * FV / 4);
        k_edge_scatter<<<edgeBlk, 256, 0, stream>>>(rows, cols, vals,
                                                    vars, cacc, NEDGES);
        k_linear32<<<(ctiles + 7) / 8, 256, 0, stream>>>(cons, cacc, Wc, bc,
                                                         cons, ctiles);
        // c -> v
        k_zero4<<<(VCOUNT * FV / 4 + 255) / 256, 256, 0, stream>>>(
            (float4*)vacc, VCOUNT * FV / 4);
        k_edge_scatter<<<edgeBlk, 256, 0, stream>>>(cols, rows, vals,
                                                    cons, vacc, NEDGES);
        k_linear32<<<(vtiles + 7) / 8, 256, 0, stream>>>(vars, vacc, Wv, bv,
                                                         vars, vtiles);
    }

    k_output<<<(vtiles + 7) / 8, 256, 0, stream>>>(vars, Wo, bo, Wo2, bo2,
                                                   outp, vtiles);
}